// MonoDepthLoss_21234318311762
// MI455X (gfx1250) — compile-verified
//
#include <hip/hip_runtime.h>
#include <hip/hip_bf16.h>
#include <math.h>

#define SSIM_C1 0.0001f
#define SSIM_C2 0.0005f
#define RED_BLOCKS 512
#define RED_THREADS 256

typedef __attribute__((ext_vector_type(2))) float v2f;
typedef __attribute__((ext_vector_type(8))) float v8f;

// ---------------------------------------------------------------------------
// Wave-level sum of 32 per-lane partials via V_WMMA_F32_16X16X4_F32.
// A: lane L (<16) -> A[L][0]=p ; lane L+16 -> A[L][2]=p ; other K cols = 0.
// B: only column N=0 is ones (lanes 0 and 16 of both B VGPRs), so
// D[m][0] = p_m + p_{m+16}, living in VGPRs 0..7 at lanes 0 and 16.
// Robust to K-row-order ambiguity in B since only N=0 is nonzero.
// EXEC must be all ones: called convergently by every thread of the block.
// ---------------------------------------------------------------------------
__device__ __forceinline__ float wave_sum(float p) {
#if __has_builtin(__builtin_amdgcn_wmma_f32_16x16x4_f32)
  unsigned lane = threadIdx.x & 31u;
  v2f a; a[0] = p; a[1] = 0.0f;
  float bsel = ((lane & 15u) == 0u) ? 1.0f : 0.0f;
  v2f b; b[0] = bsel; b[1] = bsel;
  v8f c = {};
  v8f d = __builtin_amdgcn_wmma_f32_16x16x4_f32(
      /*neg_a=*/false, a, /*neg_b=*/false, b,
      /*c_mod=*/(short)0, c, /*reuse_a=*/false, /*reuse_b=*/false);
  float s = d[0] + d[1] + d[2] + d[3] + d[4] + d[5] + d[6] + d[7];
  // lane0 holds sum of rows 0..7 of column 0, lane16 holds rows 8..15.
  float tot = s + __shfl(s, 16, 32);
  return tot;  // valid at lane 0
#else
  for (int off = 16; off > 0; off >>= 1) p += __shfl_xor(p, off, 32);
  return p;
#endif
}

__device__ __forceinline__ float block_sum(float p) {
  __shared__ float smem[RED_THREADS / 32];
  float w = wave_sum(p);
  unsigned lane = threadIdx.x & 31u;
  unsigned wid = threadIdx.x >> 5;
  if (lane == 0) smem[wid] = w;
  __syncthreads();
  float r = 0.0f;
  if (threadIdx.x == 0) {
    for (int i = 0; i < RED_THREADS / 32; ++i) r += smem[i];
  }
  return r;  // valid at thread 0
}

// ---------------------------------------------------------------------------
// Bilinear sample with zero padding (matches grid_sample_zeros corner logic).
// ---------------------------------------------------------------------------
__device__ __forceinline__ float tapz(const float* __restrict__ pl, int h, int w,
                                      int y, int x) {
  return (x >= 0 && x < w && y >= 0 && y < h) ? pl[y * w + x] : 0.0f;
}

__device__ __forceinline__ float sample_zeros(const float* __restrict__ pl, int h,
                                              int w, float ix, float iy) {
  float xf = floorf(ix), yf = floorf(iy);
  int x0 = (int)xf, y0 = (int)yf;
  float wx = ix - xf, wy = iy - yf;
  float v00 = tapz(pl, h, w, y0, x0);
  float v01 = tapz(pl, h, w, y0, x0 + 1);
  float v10 = tapz(pl, h, w, y0 + 1, x0);
  float v11 = tapz(pl, h, w, y0 + 1, x0 + 1);
  return v00 * (1.0f - wx) * (1.0f - wy) + v01 * wx * (1.0f - wy) +
         v10 * (1.0f - wx) * wy + v11 * wx * wy;
}

// ---------------------------------------------------------------------------
// Align-corners bilinear resize: (planes,H,W) -> (planes,h,w)
// ---------------------------------------------------------------------------
__global__ __launch_bounds__(256) void resize_kernel(
    const float* __restrict__ src, int H, int W, int h, int w, int planes,
    float ry, float rx, float* __restrict__ dst) {
  int N = planes * h * w;
  int idx = blockIdx.x * blockDim.x + threadIdx.x;
  if (idx >= N) return;
  int i = idx % w;
  int j = (idx / w) % h;
  int pl = idx / (w * h);
  float ys = (float)j * ry;
  float xs = (float)i * rx;
  float yf = floorf(ys), xf = floorf(xs);
  int y0 = (int)yf, x0 = (int)xf;
  int y1 = min(y0 + 1, H - 1);
  int x1 = min(x0 + 1, W - 1);
  float wy = ys - yf, wx = xs - xf;
  const float* p = src + (long long)pl * H * W;
  float t00 = p[y0 * W + x0], t01 = p[y0 * W + x1];
  float t10 = p[y1 * W + x0], t11 = p[y1 * W + x1];
  float r0 = t00 * (1.0f - wy) + t10 * wy;
  float r1 = t01 * (1.0f - wy) + t11 * wy;
  dst[idx] = r0 * (1.0f - wx) + r1 * wx;
}

// ---------------------------------------------------------------------------
// Warp src (B,3,h,w) by disparity channel `ch` (with sign) -> dst (B,3,h,w).
// ix = (i/(w-1) + d)*w - 0.5 ; iy = j*h/(h-1) - 0.5  (grid_sample reduction)
// ---------------------------------------------------------------------------
__global__ __launch_bounds__(256) void warp_kernel(
    const float* __restrict__ src, const float* __restrict__ disp, int ch,
    float sign, int B, int h, int w, float* __restrict__ dst) {
  int N = B * 3 * h * w;
  int idx = blockIdx.x * blockDim.x + threadIdx.x;
  if (idx >= N) return;
  int i = idx % w;
  int j = (idx / w) % h;
  int c = (idx / (w * h)) % 3;
  int b = idx / (3 * h * w);
  float dval = sign * disp[(long long)(b * 2 + ch) * h * w + j * w + i];
  float ix = ((float)i / (float)(w - 1) + dval) * (float)w - 0.5f;
  float iy = (float)j * (float)h / (float)(h - 1) - 0.5f;
  const float* pl = src + (long long)(b * 3 + c) * h * w;
  dst[idx] = sample_zeros(pl, h, w, ix, iy);
}

// ---------------------------------------------------------------------------
// SSIM over valid region (h-2, w-2); accumulate raw SSIM sum into partials.
// ---------------------------------------------------------------------------
__global__ __launch_bounds__(RED_THREADS) void ssim_reduce_kernel(
    const float* __restrict__ gt, const float* __restrict__ est, int planes,
    int h, int w, float* __restrict__ partials) {
  int hh = h - 2, ww = w - 2;
  int N = planes * hh * ww;
  int stride = gridDim.x * blockDim.x;
  float p = 0.0f;
  for (int idx = blockIdx.x * blockDim.x + threadIdx.x; idx < N; idx += stride) {
    int i = idx % ww;
    int j = (idx / ww) % hh;
    int pl = idx / (ww * hh);
    const float* g = gt + (long long)pl * h * w + (long long)j * w + i;
    const float* e = est + (long long)pl * h * w + (long long)j * w + i;
    float sg = 0.f, se = 0.f, sgg = 0.f, see = 0.f, sge = 0.f;
#pragma unroll
    for (int dy = 0; dy < 3; ++dy) {
#pragma unroll
      for (int dx = 0; dx < 3; ++dx) {
        float gv = g[dy * w + dx];
        float ev = e[dy * w + dx];
        sg += gv; se += ev;
        sgg += gv * gv; see += ev * ev; sge += gv * ev;
      }
    }
    const float inv9 = 1.0f / 9.0f;
    float mug = sg * inv9, mue = se * inv9;
    float sigg = sgg * inv9 - mug * mug;
    float sige = see * inv9 - mue * mue;
    float sigge = sge * inv9 - mue * mug;
    float num = (2.0f * mue * mug + SSIM_C1) * (2.0f * sigge + SSIM_C2);
    float den = (mug * mug + mue * mue + SSIM_C1) * (sigg + sige + SSIM_C2);
    p += num / den;
  }
  float r = block_sum(p);
  if (threadIdx.x == 0) partials[blockIdx.x] = r;
}

// ---------------------------------------------------------------------------
// Edge-aware smoothness: sum |dx_d|*wx/divX + |dy_d|*wy/divY (sign of disp
// channel is irrelevant under abs). img is the resized 3-channel image.
// ---------------------------------------------------------------------------
__global__ __launch_bounds__(RED_THREADS) void smooth_reduce_kernel(
    const float* __restrict__ disp, int ch, const float* __restrict__ img,
    int B, int h, int w, float invDx, float invDy, float* __restrict__ partials) {
  int N = B * h * w;
  int stride = gridDim.x * blockDim.x;
  int hw = h * w;
  float p = 0.0f;
  for (int idx = blockIdx.x * blockDim.x + threadIdx.x; idx < N; idx += stride) {
    int i = idx % w;
    int j = (idx / w) % h;
    int b = idx / hw;
    const float* dp = disp + (long long)(b * 2 + ch) * hw;
    const float* ip = img + (long long)(b * 3) * hw;
    int o = j * w + i;
    float d00 = dp[o];
    if (i < w - 1) {
      float dxd = d00 - dp[o + 1];
      float ax = fabsf(ip[o] - ip[o + 1]) + fabsf(ip[hw + o] - ip[hw + o + 1]) +
                 fabsf(ip[2 * hw + o] - ip[2 * hw + o + 1]);
      float wx = expf(-ax * (1.0f / 3.0f));
      p += fabsf(dxd) * wx * invDx;
    }
    if (j < h - 1) {
      float dyd = d00 - dp[o + w];
      float ay = fabsf(ip[o] - ip[o + w]) + fabsf(ip[hw + o] - ip[hw + o + w]) +
                 fabsf(ip[2 * hw + o] - ip[2 * hw + o + w]);
      float wyv = expf(-ay * (1.0f / 3.0f));
      p += fabsf(dyd) * wyv * invDy;
    }
  }
  float r = block_sum(p);
  if (threadIdx.x == 0) partials[blockIdx.x] = r;
}

// ---------------------------------------------------------------------------
// LR consistency, both directions in one pass:
//   dl = -disp[:,0], dr = disp[:,1]
//   term1 = |dr - warp(dr by dl)|, term2 = |dl - warp(dl by dr)|
// ---------------------------------------------------------------------------
__global__ __launch_bounds__(RED_THREADS) void lrc_reduce_kernel(
    const float* __restrict__ disp, int B, int h, int w, float invN,
    float* __restrict__ partials) {
  int N = B * h * w;
  int stride = gridDim.x * blockDim.x;
  int hw = h * w;
  float p = 0.0f;
  for (int idx = blockIdx.x * blockDim.x + threadIdx.x; idx < N; idx += stride) {
    int i = idx % w;
    int j = (idx / w) % h;
    int b = idx / hw;
    const float* d0 = disp + (long long)(b * 2) * hw;  // channel 0
    const float* d1 = d0 + hw;                         // channel 1
    int o = j * w + i;
    float dl = -d0[o];
    float dr = d1[o];
    float xb = (float)i / (float)(w - 1);
    float iy = (float)j * (float)h / (float)(h - 1) - 0.5f;
    float ix_r = (xb + dl) * (float)w - 0.5f;
    float proj_r = sample_zeros(d1, h, w, ix_r, iy);     // warp dr image by dl
    float ix_l = (xb + dr) * (float)w - 0.5f;
    float proj_l = -sample_zeros(d0, h, w, ix_l, iy);    // warp dl image by dr
    p += (fabsf(dr - proj_r) + fabsf(dl - proj_l)) * invN;
  }
  float r = block_sum(p);
  if (threadIdx.x == 0) partials[blockIdx.x] = r;
}

// ---------------------------------------------------------------------------
// Stage-2: deterministic serial sum of fixed-count partials, scaled accumulate.
// ---------------------------------------------------------------------------
__global__ void accum_kernel(const float* __restrict__ partials, int n,
                             float* __restrict__ acc, float scale) {
  if (blockIdx.x == 0 && threadIdx.x == 0) {
    float s = 0.0f;
    for (int i = 0; i < n; ++i) s += partials[i];
    acc[0] += s * scale;
  }
}

__global__ void init_acc_kernel(float* __restrict__ acc) {
  if (threadIdx.x < 3) acc[threadIdx.x] = 0.0f;
}

__global__ void finalize_kernel(const float* __restrict__ acc,
                                float* __restrict__ out) {
  if (threadIdx.x < 3) out[threadIdx.x] = acc[threadIdx.x] * 0.25f;  // /4 scales
}

// ---------------------------------------------------------------------------
// Host orchestration: 4 scales x {L,R} sides, all on `stream`, scratch in d_ws.
// ws layout (floats): [0..7] acc, [8..8+512) partials, then bufA, bufB
// bufA/bufB: B*3*H*W floats each (25.2 MB) -> total ~50.5 MB, L2-resident.
// ---------------------------------------------------------------------------
extern "C" void kernel_launch(void* const* d_in, const int* in_sizes, int n_in,
                              void* d_out, int out_size, void* d_ws,
                              size_t ws_size, hipStream_t stream) {
  const int B = 16, H = 256, W = 512;
  const float* imgL = (const float*)d_in[0];
  const float* imgR = (const float*)d_in[1];
  const float* disps[4] = {(const float*)d_in[2], (const float*)d_in[3],
                           (const float*)d_in[4], (const float*)d_in[5]};
  float* acc = (float*)d_ws;
  float* partials = acc + 8;
  float* bufA = partials + RED_BLOCKS;
  float* bufB = bufA + (size_t)B * 3 * H * W;
  float* out = (float*)d_out;

  init_acc_kernel<<<1, 32, 0, stream>>>(acc);

  for (int s = 0; s < 4; ++s) {
    int h = H >> s, w = W >> s;
    int planes = B * 3;
    int Nimg = planes * h * w;
    int mb = (Nimg + 255) / 256;
    float ry = (float)(H - 1) / (float)(h - 1);
    float rx = (float)(W - 1) / (float)(w - 1);
    float ssimScale = 1.0f / ((float)B * 3.0f * (float)(h - 2) * (float)(w - 2));
    float invDx = 1.0f / ((float)B * (float)h * (float)(w - 1));
    float invDy = 1.0f / ((float)B * (float)(h - 1) * (float)w);
    float smScale = 1.0f / ((float)((s + 1) * (s + 1)));

    for (int side = 0; side < 2; ++side) {
      const float* img = (side == 0) ? imgL : imgR;
      int ch = side;                              // dl=ch0, dr=ch1
      float sign = (side == 0) ? -1.0f : 1.0f;    // dl = -disp[:,0]

      resize_kernel<<<mb, 256, 0, stream>>>(img, H, W, h, w, planes, ry, rx,
                                            bufA);
      warp_kernel<<<mb, 256, 0, stream>>>(bufA, disps[s], ch, sign, B, h, w,
                                          bufB);
      ssim_reduce_kernel<<<RED_BLOCKS, RED_THREADS, 0, stream>>>(
          bufA, bufB, planes, h, w, partials);
      accum_kernel<<<1, 32, 0, stream>>>(partials, RED_BLOCKS, acc + 0,
                                         ssimScale);
      smooth_reduce_kernel<<<RED_BLOCKS, RED_THREADS, 0, stream>>>(
          disps[s], ch, bufA, B, h, w, invDx, invDy, partials);
      accum_kernel<<<1, 32, 0, stream>>>(partials, RED_BLOCKS, acc + 1,
                                         smScale);
    }

    float invN = 1.0f / ((float)B * (float)h * (float)w);
    lrc_reduce_kernel<<<RED_BLOCKS, RED_THREADS, 0, stream>>>(disps[s], B, h, w,
                                                              invN, partials);
    accum_kernel<<<1, 32, 0, stream>>>(partials, RED_BLOCKS, acc + 2, 1.0f);
  }

  finalize_kernel<<<1, 32, 0, stream>>>(acc, out);
}